// GRModule_90494960926988
// MI455X (gfx1250) — compile-verified
//
#include <hip/hip_runtime.h>
#include <hip/hip_bf16.h>
#include <math.h>

typedef __attribute__((ext_vector_type(2))) float v2f;
typedef __attribute__((ext_vector_type(8))) float v8f;

#define BB 16
#define CC 256
#define HW 1024
#define MM 16
#define WSTRIDE 1028   // padded LDS stride for W strip (bank spread)

static __device__ __forceinline__ v8f wmma_f32_k4(v2f a, v2f b, v8f c) {
    return __builtin_amdgcn_wmma_f32_16x16x4_f32(
        /*neg_a=*/false, a, /*neg_b=*/false, b,
        /*c_mod=*/(short)0, c, /*reuse_a=*/false, /*reuse_b=*/false);
}

// Low 32 bits of a generic shared-aperture address == LDS byte offset (ISA §10.2)
static __device__ __forceinline__ unsigned to_lds(const void* p) {
    return (unsigned)(unsigned long long)p;
}
// GLOBAL_LOAD_ASYNC_TO_LDS_B128: per-lane 16B global->LDS copy, tracked by ASYNCcnt
static __device__ __forceinline__ void async_load_b128(unsigned lds_off, const float* gaddr) {
    asm volatile("global_load_async_to_lds_b128 %0, %1, off"
                 :: "v"(lds_off), "v"(gaddr) : "memory");
}
static __device__ __forceinline__ void wait_async0() {
    asm volatile("s_wait_asynccnt 0x0" ::: "memory");
}

// ---------------- K1: xsum[b,c] = sum_p x[b,c,p] ----------------
__global__ __launch_bounds__(256) void k1_xsum(const float* __restrict__ x,
                                               float* __restrict__ xsum) {
    int row  = blockIdx.x * 8 + (threadIdx.x >> 5);   // b*256+c
    int lane = threadIdx.x & 31;
    const float* p = x + (size_t)row * HW;
    float s = 0.f;
    #pragma unroll 8
    for (int i = lane; i < HW; i += 32) {
        __builtin_prefetch(p + i + 256, 0, 0);
        s += p[i];
    }
    #pragma unroll
    for (int m = 16; m >= 1; m >>= 1) s += __shfl_xor(s, m, 32);
    if (lane == 0) xsum[row] = s;
}

// ---------------- K2: sg[b,m] = sigmoid(glob_w[m,:]·xsum[b,:]/HW) ----------------
__global__ __launch_bounds__(256) void k2_sg(const float* __restrict__ glob_w,
                                             const float* __restrict__ xsum,
                                             float* __restrict__ sg) {
    int t = threadIdx.x;            // 256 = B*M
    int b = t >> 4, m = t & 15;
    float acc = 0.f;
    for (int c = 0; c < CC; ++c) acc += glob_w[m * CC + c] * xsum[b * CC + c];
    float g = acc * (1.0f / (float)HW);
    sg[t] = 1.0f / (1.0f + __expf(-g));
}

// ---------------- K3: r = relu(phi_w @ x + phi_b), layout [B][M][HW] ----------------
__global__ __launch_bounds__(256) void k3_phi(const float* __restrict__ x,
                                              const float* __restrict__ phi_w,
                                              const float* __restrict__ phi_b,
                                              float* __restrict__ r) {
    int b    = blockIdx.x >> 3;
    int wid  = threadIdx.x >> 5;
    int p0   = ((blockIdx.x & 7) * 8 + wid) * 16;
    int lane = threadIdx.x & 31;
    int half = lane >> 4, lid = lane & 15;
    const float* xb = x + (size_t)b * CC * HW;
    v8f acc = {};
    for (int k0 = 0; k0 < CC; k0 += 4) {
        int kr = k0 + 2 * half;
        v2f a  = *(const v2f*)(phi_w + lid * CC + kr);           // A: phi_w[M=lid, K]
        v2f bv = { xb[kr * HW + p0 + lid],                       // B: x[K, N=p]
                   xb[(kr + 1) * HW + p0 + lid] };
        acc = wmma_f32_k4(a, bv, acc);
    }
    #pragma unroll
    for (int v = 0; v < 8; ++v) {
        int m = v + 8 * half;
        float val = acc[v] + phi_b[m];
        r[b * (MM * HW) + m * HW + p0 + lid] = fmaxf(val, 0.f);
    }
}

// ---------------- K4: P[b,n,j] = 0.5*rowsum_n + (sg[j]-0.5)*x_phi[n,j] ----------------
__global__ __launch_bounds__(256) void k4_P(const float* __restrict__ r,
                                            const float* __restrict__ sg,
                                            float* __restrict__ P) {
    int t = blockIdx.x * 256 + threadIdx.x;
    int j = t & 15;
    int b = t >> 14;
    float v = r[t];
    float s = v;
    #pragma unroll
    for (int m = 8; m >= 1; m >>= 1) s += __shfl_xor(s, m, 32);  // sum over j-group
    P[t] = 0.5f * s + (sg[b * MM + j] - 0.5f) * v;
}

// ---------------- K5: per-row softmax stats (max, Z) + degree d ----------------
// grid B*64 x 256 (8 waves). r[b] strip async-staged into LDS; WMMA over 64 k-tiles.
// dyn smem: Lr[16384] smx[128] ssm[128] cm[16] cz[16] dsum[16]
__global__ __launch_bounds__(256) void k5_stats(const float* __restrict__ r,
                                                const float* __restrict__ P,
                                                float* __restrict__ mx,
                                                float* __restrict__ Z,
                                                float* __restrict__ dd) {
    extern __shared__ float smem5[];
    float* Lr   = smem5;              // 16*1024
    float* smx  = smem5 + 16384;      // [8][16]
    float* ssm  = smx + 128;          // [8][16]
    float* cm   = ssm + 128;          // 16
    float* cz   = cm + 16;            // 16
    float* dsum = cz + 16;            // 16

    int b = blockIdx.x >> 6;
    int n0 = (blockIdx.x & 63) * 16;
    int tid = threadIdx.x, wid = tid >> 5, lane = tid & 31;
    int half = lane >> 4, lid = lane & 15;
    const float* rb = r + b * (MM * HW);
    const float* Pb = P + b * (HW * MM);

    // async-stage r[b] (64KB) into LDS: 4096 x 16B chunks over 256 threads
    #pragma unroll
    for (int i = 0; i < 16; ++i) {
        int chunk = tid + i * 256;
        async_load_b128(to_lds(Lr + chunk * 4), rb + chunk * 4);
    }

    v2f a[4];
    #pragma unroll
    for (int kk = 0; kk < 4; ++kk)
        a[kk] = *(const v2f*)(Pb + (n0 + lid) * MM + kk * 4 + 2 * half);

    wait_async0();
    __syncthreads();

    float rm[8], rs[8];
    #pragma unroll
    for (int v = 0; v < 8; ++v) { rm[v] = -3.4e38f; rs[v] = 0.f; }

    for (int kt = wid; kt < 64; kt += 8) {
        v8f c = {};
        #pragma unroll
        for (int kk = 0; kk < 4; ++kk) {
            int krow = kk * 4 + 2 * half;
            v2f bv = { Lr[krow * HW + kt * 16 + lid],
                       Lr[(krow + 1) * HW + kt * 16 + lid] };
            c = wmma_f32_k4(a[kk], bv, c);
        }
        #pragma unroll
        for (int v = 0; v < 8; ++v) {
            float tv = c[v];
            float tmax = tv;
            #pragma unroll
            for (int m = 8; m >= 1; m >>= 1) tmax = fmaxf(tmax, __shfl_xor(tmax, m, 32));
            float nm = fmaxf(rm[v], tmax);
            float e = __expf(tv - nm);
            float es = e;
            #pragma unroll
            for (int m = 8; m >= 1; m >>= 1) es += __shfl_xor(es, m, 32);
            rs[v] = rs[v] * __expf(rm[v] - nm) + es;
            rm[v] = nm;
        }
    }
    #pragma unroll
    for (int v = 0; v < 8; ++v) {
        int row = v + 8 * half;
        if (lid == 0) { smx[wid * 16 + row] = rm[v]; ssm[wid * 16 + row] = rs[v]; }
    }
    __syncthreads();
    if (tid < 16) {
        float gm = -3.4e38f;
        for (int w = 0; w < 8; ++w) gm = fmaxf(gm, smx[w * 16 + tid]);
        float gz = 0.f;
        for (int w = 0; w < 8; ++w) gz += ssm[w * 16 + tid] * __expf(smx[w * 16 + tid] - gm);
        cm[tid] = gm; cz[tid] = gz; dsum[tid] = 0.f;
        mx[b * HW + n0 + tid] = gm;
        Z [b * HW + n0 + tid] = gz;
    }
    __syncthreads();
    // pass 2: d = 1/sqrt(sum_k A[n,k]) (reference fidelity)
    float part[8];
    #pragma unroll
    for (int v = 0; v < 8; ++v) part[v] = 0.f;
    for (int kt = wid; kt < 64; kt += 8) {
        v8f c = {};
        #pragma unroll
        for (int kk = 0; kk < 4; ++kk) {
            int krow = kk * 4 + 2 * half;
            v2f bv = { Lr[krow * HW + kt * 16 + lid],
                       Lr[(krow + 1) * HW + kt * 16 + lid] };
            c = wmma_f32_k4(a[kk], bv, c);
        }
        #pragma unroll
        for (int v = 0; v < 8; ++v) {
            int row = v + 8 * half;
            part[v] += __expf(c[v] - cm[row]) / cz[row];
        }
    }
    #pragma unroll
    for (int v = 0; v < 8; ++v) {
        float pv = part[v];
        #pragma unroll
        for (int m = 8; m >= 1; m >>= 1) pv += __shfl_xor(pv, m, 32);
        int row = v + 8 * half;
        if (lid == 0) atomicAdd(&dsum[row], pv);
    }
    __syncthreads();
    if (tid < 16) {
        float dv = 1.0f / sqrtf(dsum[tid]);
        if (!__builtin_isfinite(dv)) dv = 0.f;
        dd[b * HW + n0 + tid] = dv;
    }
}

// ---------------- K6: out = xr - (d A d) @ xr ----------------
// grid B*32 x 512 (16 waves). Two 16-row strips per block: every xr B-operand
// load feeds two WMMA accumulators (halves L2 traffic vs one strip).
// dyn smem: Lr[16384] W[32*WSTRIDE] cm[32] cz[32] cd[32]
__global__ __launch_bounds__(512) void k6_out(const float* __restrict__ x,
                                              const float* __restrict__ r,
                                              const float* __restrict__ P,
                                              const float* __restrict__ mx,
                                              const float* __restrict__ Z,
                                              const float* __restrict__ dd,
                                              float* __restrict__ out) {
    extern __shared__ float smem[];
    float* Lr = smem;                       // 16*1024
    float* W  = smem + 16384;               // 32 * WSTRIDE
    float* cm = W + 32 * WSTRIDE;           // 32
    float* cz = cm + 32;                    // 32
    float* cd = cz + 32;                    // 32

    int b  = blockIdx.x >> 5;
    int n0 = (blockIdx.x & 31) * 32;
    int tid = threadIdx.x, wid = tid >> 5, lane = tid & 31;
    int half = lane >> 4, lid = lane & 15;
    const float* xb = x + (size_t)b * CC * HW;   // xr[k][c] = xb[k*256+c]
    const float* rb = r + b * (MM * HW);
    const float* Pb = P + b * (HW * MM);

    // async-stage r[b] (64KB): 4096 x 16B chunks over 512 threads
    #pragma unroll
    for (int i = 0; i < 8; ++i) {
        int chunk = tid + i * 512;
        async_load_b128(to_lds(Lr + chunk * 4), rb + chunk * 4);
    }

    if (tid < 32) {
        cm[tid] = mx[b * HW + n0 + tid];
        cz[tid] = Z [b * HW + n0 + tid];
        cd[tid] = dd[b * HW + n0 + tid];
    }
    v2f a2[2][4];
    #pragma unroll
    for (int s = 0; s < 2; ++s)
        #pragma unroll
        for (int kk = 0; kk < 4; ++kk)
            a2[s][kk] = *(const v2f*)(Pb + (n0 + s * 16 + lid) * MM + kk * 4 + 2 * half);

    wait_async0();
    __syncthreads();

    // Phase A: rebuild scores -> weights strip W[32][1024] in LDS
    for (int kt = wid; kt < 64; kt += 16) {
        float dk = dd[b * HW + kt * 16 + lid];
        #pragma unroll
        for (int s = 0; s < 2; ++s) {
            v8f c = {};
            #pragma unroll
            for (int kk = 0; kk < 4; ++kk) {
                int krow = kk * 4 + 2 * half;
                v2f bv = { Lr[krow * HW + kt * 16 + lid],
                           Lr[(krow + 1) * HW + kt * 16 + lid] };
                c = wmma_f32_k4(a2[s][kk], bv, c);
            }
            #pragma unroll
            for (int v = 0; v < 8; ++v) {
                int row = s * 16 + v + 8 * half;
                float w = cd[row] * (__expf(c[v] - cm[row]) / cz[row]) * dk;
                W[row * WSTRIDE + kt * 16 + lid] = w;
            }
        }
    }
    __syncthreads();

    // Phase B: two 16x16 output tiles per wave, K = 1024, shared B operand
    int c0 = wid * 16;
    v8f acc0 = {}, acc1 = {};
    for (int k0 = 0; k0 < HW; k0 += 4) {
        int kr = k0 + 2 * half;
        v2f bv  = { xb[kr * CC + c0 + lid],
                    xb[(kr + 1) * CC + c0 + lid] };
        v2f av0 = *(const v2f*)(&W[lid * WSTRIDE + kr]);
        v2f av1 = *(const v2f*)(&W[(16 + lid) * WSTRIDE + kr]);
        acc0 = wmma_f32_k4(av0, bv, acc0);
        acc1 = wmma_f32_k4(av1, bv, acc1);
    }
    #pragma unroll
    for (int v = 0; v < 8; ++v) {
        int row = v + 8 * half;
        int col = c0 + lid;
        int nA = n0 + row;
        int nB = n0 + 16 + row;
        out[((size_t)b * HW + nA) * CC + col] = xb[nA * CC + col] - acc0[v];
        out[((size_t)b * HW + nB) * CC + col] = xb[nB * CC + col] - acc1[v];
    }
}

extern "C" void kernel_launch(void* const* d_in, const int* in_sizes, int n_in,
                              void* d_out, int out_size, void* d_ws, size_t ws_size,
                              hipStream_t stream) {
    const float* x      = (const float*)d_in[0];
    const float* phi_w  = (const float*)d_in[1];
    const float* phi_b  = (const float*)d_in[2];
    const float* glob_w = (const float*)d_in[3];
    float* out = (float*)d_out;
    float* ws  = (float*)d_ws;

    float* xsum = ws;                 //  4096
    float* sg   = ws + 4096;          //   256
    float* r    = ws + 4352;          // 262144
    float* P    = ws + 266496;        // 262144
    float* mx   = ws + 528640;        // 16384
    float* Zb   = ws + 545024;        // 16384
    float* dd   = ws + 561408;        // 16384  (total ~2.3 MB)

    k1_xsum <<<512, 256, 0, stream>>>(x, xsum);
    k2_sg   <<<1,   256, 0, stream>>>(glob_w, xsum, sg);
    k3_phi  <<<128, 256, 0, stream>>>(x, phi_w, phi_b, r);
    k4_P    <<<1024,256, 0, stream>>>(r, sg, P);

    size_t shmem5 = (size_t)(16384 + 128 + 128 + 16 + 16 + 16) * sizeof(float);
    k5_stats<<<BB * 64, 256, shmem5, stream>>>(r, P, mx, Zb, dd);

    size_t shmem6 = (size_t)(16384 + 32 * WSTRIDE + 96) * sizeof(float);
    k6_out  <<<BB * 32, 512, shmem6, stream>>>(x, r, P, mx, Zb, dd, out);
}